// post_54795192762798
// MI455X (gfx1250) — compile-verified
//
#include <hip/hip_runtime.h>
#include <cstdint>

// Row softmax with argmax-keep / uniform-fill, tuned for MI455X (gfx1250):
//  - bandwidth-bound: 4 * B*C * 4B = ~1.05 GB  => ~45us floor @ 23.3 TB/s
//  - x row (128 KB) staged in LDS once via gfx1250 async global->LDS DMA
//  - u/noise streamed non-temporally, x2 kept in registers, NT stores
//  - wave32 shuffle reductions (warpSize==32 on CDNA5)

#define BLOCK  1024
#define ROWLEN 32000
#define NV4    (ROWLEN / 4)     // 8000 float4 per row
#define KMAX   8                // ceil(NV4 / BLOCK)

typedef float v4f __attribute__((ext_vector_type(4)));

__device__ __forceinline__ void wave_red_max(float& v) {
#pragma unroll
    for (int m = 16; m >= 1; m >>= 1)
        v = fmaxf(v, __shfl_xor(v, m, 32));
}

__device__ __forceinline__ void wave_red_sum(float& v) {
#pragma unroll
    for (int m = 16; m >= 1; m >>= 1)
        v += __shfl_xor(v, m, 32);
}

__device__ __forceinline__ void wave_red_argmax(float& v, int& i) {
#pragma unroll
    for (int m = 16; m >= 1; m >>= 1) {
        float ov = __shfl_xor(v, m, 32);
        int   oi = __shfl_xor(i, m, 32);
        if (ov > v || (ov == v && oi < i)) { v = ov; i = oi; }
    }
}

__global__ __launch_bounds__(BLOCK) void softmax_fill_kernel(
    const float* __restrict__ x, const float* __restrict__ u,
    const float* __restrict__ noise, float* __restrict__ out)
{
    __shared__ v4f   sXv[NV4];        // 128 KB: full row of x, then reused as-is
    __shared__ float sredv[32];
    __shared__ int   sredi[32];
    __shared__ float s_m, s_max2, s_sum;
    __shared__ int   s_idx;

    const int  tid  = threadIdx.x;
    const int  lane = tid & 31;
    const int  wid  = tid >> 5;
    const long row  = blockIdx.x;

    const float* xrow = x + row * (long)ROWLEN;
    const v4f*   urow = (const v4f*)(u     + row * (long)ROWLEN);
    const v4f*   nrow = (const v4f*)(noise + row * (long)ROWLEN);
    v4f*         orow = (v4f*)(out + row * (long)ROWLEN);

    // ---- Phase A: async DMA the x row into LDS (gfx1250 async-to-LDS path) ----
    const uint32_t ldsBase = (uint32_t)(uintptr_t)&sXv[0];
#pragma unroll
    for (int k = 0; k < KMAX; ++k) {
        int vi = tid + k * BLOCK;
        if (vi < NV4) {
            uint32_t lo = ldsBase + (uint32_t)vi * 16u;   // LDS byte address
            uint32_t go = (uint32_t)vi * 16u;             // global byte offset
            asm volatile("global_load_async_to_lds_b128 %0, %1, %2"
                         :: "v"(lo), "v"(go), "s"(xrow) : "memory");
        }
    }
    asm volatile("s_wait_asynccnt 0" ::: "memory");
    __syncthreads();

    // ---- max / argmax of x (first-occurrence tie-break, matching jnp.argmax) ----
    float mv = -INFINITY;
    int   mi = 0x7fffffff;
#pragma unroll
    for (int k = 0; k < KMAX; ++k) {
        int vi = tid + k * BLOCK;
        if (vi < NV4) {
            v4f xv = sXv[vi];
            int b = vi * 4;
            if (xv.x > mv) { mv = xv.x; mi = b;     }
            if (xv.y > mv) { mv = xv.y; mi = b + 1; }
            if (xv.z > mv) { mv = xv.z; mi = b + 2; }
            if (xv.w > mv) { mv = xv.w; mi = b + 3; }
        }
    }
    wave_red_argmax(mv, mi);
    if (lane == 0) { sredv[wid] = mv; sredi[wid] = mi; }
    __syncthreads();
    if (wid == 0) {
        float vv = sredv[lane]; int ii = sredi[lane];
        wave_red_argmax(vv, ii);
        if (lane == 0) { s_m = vv; s_idx = ii; }
    }
    __syncthreads();
    const float m    = s_m;
    const int   midx = s_idx;

    // ---- Phase B: x2 = where(keep, x, (0.1+0.2u)*m) + noise ; track max2 ----
    v4f   r[KMAX];
    float mx2 = -INFINITY;
#pragma unroll
    for (int k = 0; k < KMAX; ++k) {
        int vi = tid + k * BLOCK;
        if (vi < NV4) {
            v4f uv = __builtin_nontemporal_load(&urow[vi]);
            v4f nv = __builtin_nontemporal_load(&nrow[vi]);
            v4f xv = sXv[vi];
            int b = vi * 4;
            v4f t;
            t.x = ((b + 0) == midx ? xv.x : (0.1f + 0.2f * uv.x) * m) + nv.x;
            t.y = ((b + 1) == midx ? xv.y : (0.1f + 0.2f * uv.y) * m) + nv.y;
            t.z = ((b + 2) == midx ? xv.z : (0.1f + 0.2f * uv.z) * m) + nv.z;
            t.w = ((b + 3) == midx ? xv.w : (0.1f + 0.2f * uv.w) * m) + nv.w;
            r[k] = t;
            mx2 = fmaxf(mx2, fmaxf(fmaxf(t.x, t.y), fmaxf(t.z, t.w)));
        } else {
            r[k] = (v4f)(0.0f);
        }
    }
    wave_red_max(mx2);
    if (lane == 0) sredv[wid] = mx2;
    __syncthreads();
    if (wid == 0) { float vv = sredv[lane]; wave_red_max(vv); if (lane == 0) s_max2 = vv; }
    __syncthreads();
    const float max2 = s_max2;

    // ---- Phase C: e = exp(x2 - max2) in registers ; reduce sum ----
    float sum = 0.0f;
#pragma unroll
    for (int k = 0; k < KMAX; ++k) {
        int vi = tid + k * BLOCK;
        if (vi < NV4) {
            v4f t = r[k];
            t.x = __expf(t.x - max2);
            t.y = __expf(t.y - max2);
            t.z = __expf(t.z - max2);
            t.w = __expf(t.w - max2);
            r[k] = t;
            sum += (t.x + t.y) + (t.z + t.w);
        }
    }
    wave_red_sum(sum);
    if (lane == 0) sredv[wid] = sum;
    __syncthreads();
    if (wid == 0) { float vv = sredv[lane]; wave_red_sum(vv); if (lane == 0) s_sum = vv; }
    __syncthreads();
    const float inv = 1.0f / s_sum;

    // ---- Phase D: scale and stream out (non-temporal b128 stores) ----
#pragma unroll
    for (int k = 0; k < KMAX; ++k) {
        int vi = tid + k * BLOCK;
        if (vi < NV4) {
            v4f t = r[k];
            t.x *= inv; t.y *= inv; t.z *= inv; t.w *= inv;
            __builtin_nontemporal_store(t, &orow[vi]);
        }
    }
}

extern "C" void kernel_launch(void* const* d_in, const int* in_sizes, int n_in,
                              void* d_out, int out_size, void* d_ws, size_t ws_size,
                              hipStream_t stream) {
    (void)n_in; (void)d_ws; (void)ws_size; (void)out_size;
    const float* x     = (const float*)d_in[0];
    const float* u     = (const float*)d_in[1];
    const float* noise = (const float*)d_in[2];
    float*       out   = (float*)d_out;

    const int rows = in_sizes[0] / ROWLEN;   // 2048
    softmax_fill_kernel<<<dim3(rows), dim3(BLOCK), 0, stream>>>(x, u, noise, out);
}